// LSTMTagger_CRF_52235392254600
// MI455X (gfx1250) — compile-verified
//
#include <hip/hip_runtime.h>
#include <stdint.h>

// Viterbi CRF decode for MI455X (gfx1250), round 2.
// - One workgroup per batch; backpointers LDS-resident (no bp HBM traffic).
// - feats streamed by the CDNA5 Tensor Data Mover (tensor_load_to_lds),
//   double-buffered, synchronized with s_wait_tensorcnt + barriers.
// - Masked (t >= length) steps are skipped entirely: prefix mask means the
//   reference's frozen-score/identity-bp steps are no-ops for the result.
//   Average length = ~T/2 -> ~2x less compute and ~2x less feats traffic.
// - trans kept TRANSPOSED in LDS (transT[j][i] = trans[i][j], stride 130):
//   lane j reads contiguous float2 in i, bank-conflict-free (2j mod 64),
//   8B-aligned rows; score reads are uniform float2 broadcasts.
// - Max-plus + argmax cannot map to WMMA (not an IEEE x/+ reduction).
// Assumptions: masks arrive as int32; outputs written as float.

#define K_TAGS   130
#define START_T  128
#define STOP_T   129
#define T_LEN    200
#define TB       10                 // time steps per TDM tile
#define NT       (T_LEN / TB)       // 20 tiles
#define BLOCK    160                // 5 wave32
#define NEG_INF  (-3.402823466e38f)

typedef unsigned int u32x4 __attribute__((ext_vector_type(4)));
typedef int          i32x4 __attribute__((ext_vector_type(4)));
typedef int          i32x8 __attribute__((ext_vector_type(8)));

__device__ __forceinline__ uint32_t lds_byte_offset(const void* p) {
  // Flat shared-aperture addresses carry the LDS byte offset in the low 32 bits.
  return (uint32_t)(uintptr_t)p;
}

// One TDM 2D-tile load (f32) from global into LDS (CDNA5 ISA ch.8 descriptor).
__device__ __forceinline__ void tdm_load_tile_f32(uint32_t lds_off,
                                                  const float* gptr,
                                                  uint32_t tensor_d0,
                                                  uint32_t tensor_d1,
                                                  uint32_t tile_d0,
                                                  uint32_t tile_d1,
                                                  uint64_t d0_stride_elems) {
  uint64_t ga = (uint64_t)(uintptr_t)gptr;
  u32x4 g0;
  g0.x = 1u;                                               // count=1 (valid)
  g0.y = lds_off;                                          // LDS byte address
  g0.z = (uint32_t)ga;                                     // global_addr[31:0]
  g0.w = ((uint32_t)(ga >> 32) & 0x1FFFFFFu) | (2u << 30); // addr[56:32], type=2
  i32x8 g1;
  g1[0] = (int)(2u << 16);                                 // data_size=2 -> 4B
  g1[1] = (int)((tensor_d0 & 0xFFFFu) << 16);
  g1[2] = (int)((tensor_d0 >> 16) | ((tensor_d1 & 0xFFFFu) << 16));
  g1[3] = (int)((tensor_d1 >> 16) | ((tile_d0 & 0xFFFFu) << 16));
  g1[4] = (int)(tile_d1 & 0xFFFFu);                        // tile_dim2 = 0
  g1[5] = (int)(uint32_t)d0_stride_elems;
  g1[6] = (int)((uint32_t)(d0_stride_elems >> 32) & 0xFFFFu);
  g1[7] = 0;
  i32x4 gz = (i32x4)0;
#if defined(__clang_major__) && (__clang_major__ >= 23)
  i32x8 gz8 = (i32x8)0;
  __builtin_amdgcn_tensor_load_to_lds(g0, g1, gz, gz, gz8, 0);
#else
  __builtin_amdgcn_tensor_load_to_lds(g0, g1, gz, gz, 0);
#endif
}

__global__ __launch_bounds__(BLOCK)
void viterbi_crf_kernel(const float* __restrict__ feats,
                        const int*   __restrict__ masks,
                        const float* __restrict__ trans,
                        float* __restrict__ out, int B) {
  extern __shared__ unsigned char smem[];
  float*   s_transT = (float*)smem;                  // K*K transposed (67600 B)
  float*   s_emit   = s_transT + K_TAGS * K_TAGS;    // 2*TB*K        (10400 B)
  float*   s_score  = s_emit + 2 * TB * K_TAGS;      // 2*K           (1040 B)
  float*   s_red    = s_score + 2 * K_TAGS;          // BLOCK         (640 B)
  int*     s_len    = (int*)(s_red + BLOCK);         // 1             (4 B)
  uint8_t* s_path   = (uint8_t*)(s_len + 1);         // T             (200 B)
  uint8_t* s_bp     = s_path + T_LEN;                // T*K           (26000 B)

  const int b   = blockIdx.x;
  const int tid = threadIdx.x;
  const int j   = tid;
  const bool wave0 = (tid < 32);                     // wave-uniform predicate

  const float* fb = feats + (size_t)b * T_LEN * K_TAGS;

  if (tid == 0) *s_len = 0;
  __syncthreads();

  // Wave0 launches the first two feats tiles on the TDM; meanwhile all
  // threads count the mask prefix and build the transposed trans in LDS.
  if (wave0) {
    tdm_load_tile_f32(lds_byte_offset(s_emit), fb,
                      K_TAGS, TB, K_TAGS, TB, K_TAGS);
    tdm_load_tile_f32(lds_byte_offset(s_emit + TB * K_TAGS),
                      fb + (size_t)TB * K_TAGS,
                      K_TAGS, TB, K_TAGS, TB, K_TAGS);
  }
  {
    int cnt = 0;
    for (int t = tid; t < T_LEN; t += BLOCK)
      cnt += (masks[(size_t)b * T_LEN + t] != 0) ? 1 : 0;
    atomicAdd(s_len, cnt);
  }
  if (j < K_TAGS) {
    // Coalesced global reads (row i, lanes j consecutive); conflict-free
    // LDS writes (dword j*130+i -> bank 2j mod 64, distinct per lane).
    for (int i = 0; i < K_TAGS; ++i)
      s_transT[j * K_TAGS + i] = trans[i * K_TAGS + j];
  }
  if (wave0) __builtin_amdgcn_s_wait_tensorcnt(1);   // tile0 done (in-order)
  __syncthreads();

  const int L = *s_len;                              // sequence length (1..T)

  // t = 0 init: score0[j] = trans[START, j] + feats[b, 0, j]
  int cur = 0;
  if (j < K_TAGS) s_score[j] = s_transT[j * K_TAGS + START_T] + s_emit[j];
  __syncthreads();

  int buf = 0;
  for (int tile = 0; tile < NT && tile * TB < L; ++tile) {
    const int t0    = tile * TB;
    const int t_end = (t0 + TB < L) ? (t0 + TB) : L;
    const float* ebuf = s_emit + buf * (TB * K_TAGS);

    for (int t = (t0 == 0 ? 1 : t0); t < t_end; ++t) {
      if (j < K_TAGS) {
        const float*  sc  = s_score + cur * K_TAGS;
        const float2* sc2 = (const float2*)sc;
        const float2* tr2 = (const float2*)(s_transT + j * K_TAGS);
        float best = NEG_INF;
        int   bidx = 0;
        #pragma unroll 8
        for (int h = 0; h < K_TAGS / 2; ++h) {       // ds_load_b64 pairs
          float2 tv = tr2[h];
          float2 sv = sc2[h];
          float c0 = sv.x + tv.x;
          float c1 = sv.y + tv.y;
          if (c0 > best) { best = c0; bidx = 2 * h; }     // '>' keeps first tie
          if (c1 > best) { best = c1; bidx = 2 * h + 1; }
        }
        best += ebuf[(t - t0) * K_TAGS + j];
        s_score[(cur ^ 1) * K_TAGS + j] = best;      // t < L -> always unmasked
        s_bp[t * K_TAGS + j] = (uint8_t)bidx;
      }
      cur ^= 1;
      __syncthreads();
    }

    // Refill the consumed buffer with tile+2 if it's still inside [0, L).
    if (wave0) {
      const bool issue = (tile + 2 < NT) && ((tile + 2) * TB < L);
      if (issue) {
        tdm_load_tile_f32(lds_byte_offset(s_emit + buf * (TB * K_TAGS)),
                          fb + (size_t)(tile + 2) * TB * K_TAGS,
                          K_TAGS, TB, K_TAGS, TB, K_TAGS);
        __builtin_amdgcn_s_wait_tensorcnt(1);
      } else {
        __builtin_amdgcn_s_wait_tensorcnt(0);
      }
    }
    buf ^= 1;
    __syncthreads();
  }

  // Termination: end[j] = score[j] + trans[j, STOP] = score[j] + transT[STOP][j].
  float endv = NEG_INF;
  if (j < K_TAGS) endv = s_score[cur * K_TAGS + j] + s_transT[STOP_T * K_TAGS + j];
  s_red[j] = endv;
  __syncthreads();

  if (tid == 0) {
    float bs = s_red[0]; int bt = 0;
    for (int jj = 1; jj < K_TAGS; ++jj) {
      float v = s_red[jj];
      if (v > bs) { bs = v; bt = jj; }
    }
    out[b] = bs;                                     // path_score
    int tag = bt;                                    // backtrace, LDS only
    for (int t = L - 1; t >= 1; --t) {
      s_path[t] = (uint8_t)tag;
      tag = s_bp[t * K_TAGS + tag];
    }
    s_path[0] = (uint8_t)tag;
  }
  __syncthreads();

  float* pout = out + B + (size_t)b * T_LEN;
  for (int t = tid; t < T_LEN; t += BLOCK)
    pout[t] = (t < L) ? (float)s_path[t] : 0.0f;     // masked tail -> 0
}

extern "C" void kernel_launch(void* const* d_in, const int* in_sizes, int n_in,
                              void* d_out, int out_size, void* d_ws, size_t ws_size,
                              hipStream_t stream) {
  const float* feats = (const float*)d_in[0];
  const int*   masks = (const int*)d_in[1];   // jnp bool -> assumed int32 per harness
  const float* trans = (const float*)d_in[2];
  float* out = (float*)d_out;                 // [B] path_score, then [B*T] path (f32)

  const int B = in_sizes[1] / T_LEN;
  const size_t smem =
      (size_t)(K_TAGS * K_TAGS + 2 * TB * K_TAGS + 2 * K_TAGS + BLOCK) * sizeof(float) +
      sizeof(int) + (size_t)(T_LEN + T_LEN * K_TAGS);

  // >64KB dynamic LDS: raise the limit (CDNA5 WGP has 320KB).
  (void)hipFuncSetAttribute((const void*)viterbi_crf_kernel,
                            hipFuncAttributeMaxDynamicSharedMemorySize, (int)smem);

  viterbi_crf_kernel<<<B, BLOCK, smem, stream>>>(feats, masks, trans, out, B);
}